// Attention_1503238554188
// MI455X (gfx1250) — compile-verified
//
#include <hip/hip_runtime.h>
#include <hip/hip_bf16.h>

typedef __attribute__((ext_vector_type(2))) float v2f;
typedef __attribute__((ext_vector_type(8))) float v8f;

// ---------------- per-wave LDS scratch layout (floats) ----------------
#define HOFF   0      // H tile, 16 x 44 (K padded 41->44, zeros)
#define EOFF   704    // E,     16 x 36 (32 cols, stride 36 for bank spread)
#define QOFF   1280   // Q (later reused for G), 16 x 68
#define KOFF   2368   // K,     16 x 68
#define VOFF   3456   // V^T,   64 x 18
#define AOFF   4608   // attn,  16 x 20
#define WSZ    4928   // floats per-wave scratch

// ---------------- weight fragment area (floats) ----------------
#define W1F    0      // 11 steps * 2 tiles * 64 = 1408
#define W2F    1408   //  8 * 2 * 64            = 1024
#define WQF    2432   //  8 * 4 * 64            = 2048
#define WKF    4480
#define WVF    6528
#define WOF    8576   // 16 * 2 * 64            = 2048
#define WTOT   10624

#define WAVES_PER_BLOCK 4
#define SMEM_FLOATS (WTOT + WAVES_PER_BLOCK * WSZ)   // 30336 floats = 121,344 B

static __device__ __forceinline__ v8f wmma4(v2f a, v2f b, v8f c) {
  // V_WMMA_F32_16X16X4_F32 : D = A(16x4,f32) * B(4x16,f32) + C(16x16,f32)
  return __builtin_amdgcn_wmma_f32_16x16x4_f32(
      /*neg_a=*/false, a, /*neg_b=*/false, b,
      /*c_mod=*/(short)0, c, /*reuse_a=*/false, /*reuse_b=*/false);
}

// Stage W (row-major [Nrows, Kact]) as B-fragments of W^T (B[k][n] = W[n][k]).
// Fragment storage: dst[(((s*NT)+t)*32 + lane)*2 + j]
// lane<16 holds K = s*4+j ; lane>=16 holds K = s*4+2+j ; n = t*16 + (lane&15)
template <int STEPS, int NT>
static __device__ __forceinline__ void stage_wfrag(float* dst, const float* W,
                                                   int Kact, int tid, int nth) {
  const int total = STEPS * NT * 64;
  for (int idx = tid; idx < total; idx += nth) {
    int j = idx & 1;
    int l = (idx >> 1) & 31;
    int st = idx >> 6;
    int t = st % NT;
    int s = st / NT;
    int k = s * 4 + j + ((l & 16) ? 2 : 0);
    int n = t * 16 + (l & 15);
    dst[idx] = (k < Kact) ? W[n * Kact + k] : 0.0f;
  }
}

// GEMM: A (row-major [16, astride] in LDS) x pre-swizzled weight fragments.
template <int STEPS, int NT>
static __device__ __forceinline__ void gemm_w(const float* A, int astride,
                                              const float* wf, int lane,
                                              v8f acc[NT]) {
  const int lm = lane & 15;
  const int ho = (lane & 16) ? 2 : 0;
#pragma unroll
  for (int s = 0; s < STEPS; ++s) {
    v2f a = *(const v2f*)(A + lm * astride + s * 4 + ho);
#pragma unroll
    for (int t = 0; t < NT; ++t) {
      v2f b = *(const v2f*)(wf + ((s * NT + t) * 32 + lane) * 2);
      acc[t] = wmma4(a, b, acc[t]);
    }
  }
}

__global__ __launch_bounds__(WAVES_PER_BLOCK * 32, 1) void attn_kernel(
    const float* __restrict__ H,  const float* __restrict__ W1,
    const float* __restrict__ b1, const float* __restrict__ W2,
    const float* __restrict__ b2, const float* __restrict__ Wq,
    const float* __restrict__ Wk, const float* __restrict__ Wv,
    const float* __restrict__ Wo, const float* __restrict__ bo,
    float* __restrict__ Gout, float* __restrict__ Aout, int Bn) {
  __shared__ float smem[SMEM_FLOATS];

  const int tid = threadIdx.x;
  const int nth = blockDim.x;

  // ---- stage all weights into WMMA B-fragment layout (once per block) ----
  stage_wfrag<11, 2>(smem + W1F, W1, 41, tid, nth);
  stage_wfrag<8, 2>(smem + W2F, W2, 32, tid, nth);
  stage_wfrag<8, 4>(smem + WQF, Wq, 32, tid, nth);
  stage_wfrag<8, 4>(smem + WKF, Wk, 32, tid, nth);
  stage_wfrag<8, 4>(smem + WVF, Wv, 32, tid, nth);
  stage_wfrag<16, 2>(smem + WOF, Wo, 64, tid, nth);
  __syncthreads();

  const int lane = tid & 31;
  const int lm = lane & 15;
  const int ho = (lane & 16) ? 2 : 0;  // A/B fragment K sub-offset
  const int h8 = (lane & 16) ? 8 : 0;  // C/D fragment row offset
  float* ws = smem + WTOT + (tid >> 5) * WSZ;

  // biases per output tile (n = t*16 + lm)
  float bias1[2] = {b1[lm], b1[16 + lm]};
  float bias2[2] = {b2[lm], b2[16 + lm]};
  float biasO[2] = {bo[lm], bo[16 + lm]};

  const v8f zero8 = {0.f, 0.f, 0.f, 0.f, 0.f, 0.f, 0.f, 0.f};

  const int gw = (blockIdx.x * blockDim.x + tid) >> 5;
  const int nw = (gridDim.x * blockDim.x) >> 5;

  for (int b = gw; b < Bn; b += nw) {
    // ---- prefetch next batch's H tile into cache (global_prefetch_b8);
    //      84B lane stride covers every 128B line of the 2624B tile ----
    {
      int bn = b + nw;
      if (bn < Bn) {
        const float* Hn = H + (size_t)bn * 656 + lane * 21;
        __builtin_prefetch(Hn, 0, 0);
      }
    }

    // ---- stage H[b] (16x41 contiguous) into padded [16][44] LDS ----
    const float* Hb = H + (size_t)b * 656;
    for (int i = lane; i < 656; i += 32) {
      int r = i / 41;
      int c = i - r * 41;
      ws[HOFF + r * 44 + c] = Hb[i];
    }
    for (int i = lane; i < 48; i += 32) {  // zero K-pad cols 41..43
      int r = i / 3;
      int c = i - r * 3;
      ws[HOFF + r * 44 + 41 + c] = 0.0f;
    }

    // ---- layer 1: E = relu(H @ W1^T + b1), 16x32 ----
    {
      v8f acc[2] = {zero8, zero8};
      gemm_w<11, 2>(ws + HOFF, 44, smem + W1F, lane, acc);
#pragma unroll
      for (int t = 0; t < 2; ++t)
#pragma unroll
        for (int v = 0; v < 8; ++v) {
          float x = fmaxf(acc[t][v] + bias1[t], 0.0f);
          ws[EOFF + (v + h8) * 36 + t * 16 + lm] = x;
        }
    }
    // ---- layer 2: E = relu(E @ W2^T + b2) ----
    {
      v8f acc[2] = {zero8, zero8};
      gemm_w<8, 2>(ws + EOFF, 36, smem + W2F, lane, acc);
#pragma unroll
      for (int t = 0; t < 2; ++t)
#pragma unroll
        for (int v = 0; v < 8; ++v) {
          float x = fmaxf(acc[t][v] + bias2[t], 0.0f);
          ws[EOFF + (v + h8) * 36 + t * 16 + lm] = x;
        }
    }
    // ---- Q = E @ Wq^T (16x64), row-major stride 68 ----
    {
      v8f acc[4] = {zero8, zero8, zero8, zero8};
      gemm_w<8, 4>(ws + EOFF, 36, smem + WQF, lane, acc);
#pragma unroll
      for (int t = 0; t < 4; ++t)
#pragma unroll
        for (int v = 0; v < 8; ++v)
          ws[QOFF + (v + h8) * 68 + t * 16 + lm] = acc[t][v];
    }
    // ---- K = E @ Wk^T (16x64), row-major stride 68 ----
    {
      v8f acc[4] = {zero8, zero8, zero8, zero8};
      gemm_w<8, 4>(ws + EOFF, 36, smem + WKF, lane, acc);
#pragma unroll
      for (int t = 0; t < 4; ++t)
#pragma unroll
        for (int v = 0; v < 8; ++v)
          ws[KOFF + (v + h8) * 68 + t * 16 + lm] = acc[t][v];
    }
    // ---- V = E @ Wv^T, stored TRANSPOSED: Vt[d][m], stride 18 ----
    {
      v8f acc[4] = {zero8, zero8, zero8, zero8};
      gemm_w<8, 4>(ws + EOFF, 36, smem + WVF, lane, acc);
#pragma unroll
      for (int t = 0; t < 4; ++t)
#pragma unroll
        for (int v = 0; v < 8; ++v)
          ws[VOFF + (t * 16 + lm) * 18 + v + h8] = acc[t][v];
    }
    // ---- scores = Q @ K^T (16x16, K-dim 64); B[k][n]=K[n][k] -> same
    //      addressing as A-from-Q, just from the K buffer ----
    v8f sc = zero8;
#pragma unroll
    for (int s = 0; s < 16; ++s) {
      v2f a = *(const v2f*)(ws + QOFF + lm * 68 + s * 4 + ho);
      v2f bb = *(const v2f*)(ws + KOFF + lm * 68 + s * 4 + ho);
      sc = wmma4(a, bb, sc);
    }
    // ---- scale, diag mask, softmax over N (16 lanes per row) ----
    float av[8];
#pragma unroll
    for (int v = 0; v < 8; ++v) {
      float x = sc[v] * 0.125f;           // 1/sqrt(64)
      if (lm == v + h8) x = -1e30f;       // mask self-attention diag
      float m = x;
#pragma unroll
      for (int off = 8; off; off >>= 1) m = fmaxf(m, __shfl_xor(m, off, 32));
      float e = __expf(x - m);
      float s = e;
#pragma unroll
      for (int off = 8; off; off >>= 1) s += __shfl_xor(s, off, 32);
      av[v] = e / s;
    }
#pragma unroll
    for (int v = 0; v < 8; ++v) {
      int M = v + h8;
      ws[AOFF + M * 20 + lm] = av[v];
      Aout[(size_t)b * 256 + M * 16 + lm] = av[v];
    }
    // ---- G = A @ V (16x64); B[k][n] = V[k][n] = Vt[n][k] ----
    {
      v8f acc[4] = {zero8, zero8, zero8, zero8};
#pragma unroll
      for (int s = 0; s < 4; ++s) {
        v2f a = *(const v2f*)(ws + AOFF + lm * 20 + s * 4 + ho);
#pragma unroll
        for (int t = 0; t < 4; ++t) {
          v2f bv = *(const v2f*)(ws + VOFF + (t * 16 + lm) * 18 + s * 4 + ho);
          acc[t] = wmma4(a, bv, acc[t]);
        }
      }
#pragma unroll
      for (int t = 0; t < 4; ++t)  // reuse Q buffer as G, stride 68
#pragma unroll
        for (int v = 0; v < 8; ++v)
          ws[QOFF + (v + h8) * 68 + t * 16 + lm] = acc[t][v];
    }
    // ---- out = relu(G @ Wo^T + bo), 16x32, straight to global ----
    {
      v8f acc[2] = {zero8, zero8};
      gemm_w<16, 2>(ws + QOFF, 68, smem + WOF, lane, acc);
#pragma unroll
      for (int t = 0; t < 2; ++t)
#pragma unroll
        for (int v = 0; v < 8; ++v) {
          float x = fmaxf(acc[t][v] + biasO[t], 0.0f);
          Gout[(size_t)b * 512 + (v + h8) * 32 + t * 16 + lm] = x;
        }
    }
  }
}

extern "C" void kernel_launch(void* const* d_in, const int* in_sizes, int n_in,
                              void* d_out, int out_size, void* d_ws,
                              size_t ws_size, hipStream_t stream) {
  const float* H = (const float*)d_in[0];
  const float* W1 = (const float*)d_in[1];
  const float* b1 = (const float*)d_in[2];
  const float* W2 = (const float*)d_in[3];
  const float* b2 = (const float*)d_in[4];
  const float* Wq = (const float*)d_in[5];
  const float* Wk = (const float*)d_in[6];
  const float* Wv = (const float*)d_in[7];
  const float* Wo = (const float*)d_in[8];
  const float* bo = (const float*)d_in[9];

  int Bn = in_sizes[0] / 656;  // B * 16 * 41 elements in H
  float* Gout = (float*)d_out;
  float* Aout = Gout + (size_t)Bn * 512;

  dim3 grid(2048), block(WAVES_PER_BLOCK * 32);
  attn_kernel<<<grid, block, 0, stream>>>(H, W1, b1, W2, b2, Wq, Wk, Wv, Wo,
                                          bo, Gout, Aout, Bn);
}